// Attention_60189671686666
// MI455X (gfx1250) — compile-verified
//
#include <hip/hip_runtime.h>
#include <hip/hip_bf16.h>

typedef float v2f __attribute__((ext_vector_type(2)));
typedef float v8f __attribute__((ext_vector_type(8)));

#define BDIM 32
#define NDIM 128
#define HDIM 256

// LDS layout for k_score: Y[b] staged as 128 rows x 260-float pitch
// (pitch*4 = 1040 B: 16B-aligned rows, bank-stride 4 => conflict-light
//  j-indexed b128 reads), then w_att (256 floats).
#define PITCH   260
#define SWOFF   (NDIM * PITCH)                  // float offset of w_att
#define LDSB    ((SWOFF + HDIM) * 4)            // 134,144 bytes (< 320KB WGP LDS)

// ---------------------------------------------------------------------------
// Kernel A: Y = X @ W_pair      X:(B*N, H) row-major, W:(H,H) row-major
// One wave per 16x16 output tile, V_WMMA_F32_16X16X4_F32, K-loop of 64 steps.
// ---------------------------------------------------------------------------
__global__ __launch_bounds__(128) void k_gemm_xw(const float* __restrict__ X,
                                                 const float* __restrict__ Wp,
                                                 float* __restrict__ Y) {
  const int wave = (blockIdx.x * blockDim.x + threadIdx.x) >> 5;  // 0..4095
  const int lane = threadIdx.x & 31;
  const int half = lane >> 4;   // 0: K 0/1, 1: K 2/3
  const int l16  = lane & 15;
  const int tm = wave >> 4;     // 0..255  (M / 16)
  const int tn = wave & 15;     // 0..15   (N / 16)
  const int M0 = tm * 16, N0 = tn * 16;

  v8f acc = {};
  const float* arow = X  + (M0 + l16) * HDIM;   // A: lane holds row M0+l16
  const float* bcol = Wp + N0 + l16;            // B: lane holds col N0+l16

  for (int k0 = 0; k0 < HDIM; k0 += 4) {
    const int ka = k0 + 2 * half;
    v2f a, b;
    a.x = arow[ka];
    a.y = arow[ka + 1];
    b.x = bcol[(size_t)ka * HDIM];
    b.y = bcol[(size_t)(ka + 1) * HDIM];
    acc = __builtin_amdgcn_wmma_f32_16x16x4_f32(false, a, false, b,
                                                (short)0, acc, false, false);
  }

#pragma unroll
  for (int r = 0; r < 8; ++r) {
    const int row = M0 + r + 8 * half;          // C/D layout: M = r + 8*half
    Y[(size_t)row * HDIM + N0 + l16] = acc[r];
  }
}

// ---------------------------------------------------------------------------
// Kernel B: score[b,i,j] = sigmoid( sum_h relu(Y[b,i,h]+Y[b,j,h]) * w[h] + b0 )
// One block per (b, 16-row i-tile): 256 blocks, 256 threads.
// Whole Y[b] (128KB) staged into LDS via async global->LDS b128 copies
// (ASYNCcnt), then all 2048 (i,j) dots served from LDS with b128 reads.
// ---------------------------------------------------------------------------
__global__ __launch_bounds__(256) void k_score(const float* __restrict__ Y,
                                               const float* __restrict__ Watt,
                                               const float* __restrict__ batt,
                                               float* __restrict__ S) {
  extern __shared__ float smem[];                 // dynamic LDS, base offset 0
  const int b     = blockIdx.x >> 3;              // 0..31
  const int itile = blockIdx.x & 7;               // 0..7 (16 i-rows each)
  const int tid   = threadIdx.x;

  const float* Yb = Y + (size_t)b * NDIM * HDIM;
  __builtin_prefetch(Yb + (tid << 2), 0, 1);      // global_prefetch_b8

  // --- async stage Y[b]: 128 rows x 64 x 16B chunks = 8192 chunks ---------
  for (int c = tid; c < NDIM * (HDIM / 4); c += 256) {
    const int row = c >> 6;                       // 0..127
    const int col = c & 63;                       // 16B chunk within row
    const unsigned lds_off = (unsigned)(row * (PITCH * 4) + col * 16);
    const float*   gp      = Yb + row * HDIM + col * 4;
    asm volatile("global_load_async_to_lds_b128 %0, %1, off"
                 :: "v"(lds_off), "v"(gp) : "memory");
  }
  // --- async stage w_att: 256 floats = 64 chunks --------------------------
  if (tid < HDIM / 4) {
    const unsigned lds_off = (unsigned)(SWOFF * 4 + tid * 16);
    const float*   gp      = Watt + tid * 4;
    asm volatile("global_load_async_to_lds_b128 %0, %1, off"
                 :: "v"(lds_off), "v"(gp) : "memory");
  }
  asm volatile("s_wait_asynccnt 0x0" ::: "memory");
  __syncthreads();

  const int   j     = tid & 127;                  // second atom index
  const int   ihalf = tid >> 7;                   // 0..1
  const float bias  = batt[0];

  for (int ii = 0; ii < 8; ++ii) {
    const int i = itile * 16 + ii * 2 + ihalf;    // row of this pair
    float s = 0.f;
#pragma unroll 4
    for (int h = 0; h < HDIM; h += 4) {
      const float4 yi = *(const float4*)&smem[i * PITCH + h];   // broadcast
      const float4 yj = *(const float4*)&smem[j * PITCH + h];
      const float4 w  = *(const float4*)&smem[SWOFF + h];
      float v0 = yi.x + yj.x; v0 = v0 > 0.f ? v0 : 0.f;
      float v1 = yi.y + yj.y; v1 = v1 > 0.f ? v1 : 0.f;
      float v2 = yi.z + yj.z; v2 = v2 > 0.f ? v2 : 0.f;
      float v3 = yi.w + yj.w; v3 = v3 > 0.f ? v3 : 0.f;
      s = fmaf(v0, w.x, s);
      s = fmaf(v1, w.y, s);
      s = fmaf(v2, w.z, s);
      s = fmaf(v3, w.w, s);
    }
    const float t = s + bias;
    S[((size_t)b * NDIM + i) * NDIM + j] = 1.0f / (1.0f + __expf(-t));
  }
}

// ---------------------------------------------------------------------------
// Kernel C: G[b] = S[b] @ X[b]    S:(N,N), X:(N,H), per-batch GEMM via WMMA.
// One wave per 16x16 tile: 32 batches * 8 * 16 tiles = 4096 waves.
// ---------------------------------------------------------------------------
__global__ __launch_bounds__(128) void k_gemm_sx(const float* __restrict__ S,
                                                 const float* __restrict__ X,
                                                 float* __restrict__ G) {
  const int wave = (blockIdx.x * blockDim.x + threadIdx.x) >> 5;  // 0..4095
  const int lane = threadIdx.x & 31;
  const int half = lane >> 4;
  const int l16  = lane & 15;
  const int b  = wave >> 7;            // /128 tiles per batch
  const int t  = wave & 127;
  const int tm = t >> 4;               // 0..7
  const int tn = t & 15;               // 0..15
  const int M0 = tm * 16, N0 = tn * 16;

  const float* Sb = S + (size_t)b * NDIM * NDIM;
  const float* Xb = X + (size_t)b * NDIM * HDIM;
  float*       Gb = G + (size_t)b * NDIM * HDIM;

  v8f acc = {};
  const float* arow = Sb + (M0 + l16) * NDIM;
  const float* bcol = Xb + N0 + l16;

  for (int k0 = 0; k0 < NDIM; k0 += 4) {
    const int ka = k0 + 2 * half;
    v2f a, bf;
    a.x  = arow[ka];
    a.y  = arow[ka + 1];
    bf.x = bcol[(size_t)ka * HDIM];
    bf.y = bcol[(size_t)(ka + 1) * HDIM];
    acc = __builtin_amdgcn_wmma_f32_16x16x4_f32(false, a, false, bf,
                                                (short)0, acc, false, false);
  }

#pragma unroll
  for (int r = 0; r < 8; ++r) {
    Gb[(size_t)(M0 + r + 8 * half) * HDIM + N0 + l16] = acc[r];
  }
}

// ---------------------------------------------------------------------------
extern "C" void kernel_launch(void* const* d_in, const int* in_sizes, int n_in,
                              void* d_out, int out_size, void* d_ws, size_t ws_size,
                              hipStream_t stream) {
  const float* X    = (const float*)d_in[0];   // local_feats (B,N,H) f32
  // d_in[1] = sparse_idx (int64) — unused by the reference forward
  const float* Wp   = (const float*)d_in[2];   // W_pair (H,H) f32
  const float* Watt = (const float*)d_in[3];   // W_att  (H,1) f32
  const float* batt = (const float*)d_in[4];   // b_att  (1,)  f32
  float*       out  = (float*)d_out;           // (B,N,H) f32

  // Workspace: Y = X@W_pair (B*N*H floats = 4MB), then scores (B*N*N = 2MB)
  float* Y = (float*)d_ws;
  float* S = Y + (size_t)BDIM * NDIM * HDIM;

  // A: 4096 tiles, 4 waves/block -> 1024 blocks
  k_gemm_xw<<<1024, 128, 0, stream>>>(X, Wp, Y);
  // B: one block per (b, i-tile); whole Y[b] staged in 134KB of LDS
  k_score<<<BDIM * (NDIM / 16), 256, LDSB, stream>>>(Y, Watt, batt, S);
  // C: 4096 tiles, 4 waves/block -> 1024 blocks
  k_gemm_sx<<<1024, 128, 0, stream>>>(S, X, out);
}